// GAT_74775380624058
// MI455X (gfx1250) — compile-verified
//
#include <hip/hip_runtime.h>
#include <math.h>

typedef __attribute__((ext_vector_type(2))) float v2f;
typedef __attribute__((ext_vector_type(8))) float v8f;

#define NEG_SLOPE 0.2f

__device__ __forceinline__ float lrelu(float v) { return v > 0.0f ? v : NEG_SLOPE * v; }

// Correct float atomic max for signed floats:
// non-negative -> int max (int order == float order); negative -> uint min.
__device__ __forceinline__ void atomicMaxF(float* addr, float v) {
    if (v >= 0.0f) atomicMax((int*)addr, __float_as_int(v));
    else           atomicMin((unsigned int*)addr, __float_as_uint(v));
}

__device__ __forceinline__ void edge_sd(const long long* __restrict__ ei, int E0, int e,
                                        int* s, int* d) {
    if (e < E0) { *s = (int)ei[e]; *d = (int)ei[(size_t)E0 + e]; }
    else        { int n = e - E0; *s = n; *d = n; }
}

// ---------------------------------------------------------------------------
// generic fill
// ---------------------------------------------------------------------------
__global__ void fill_kernel(float* __restrict__ p, float v, size_t n) {
    size_t i = (size_t)blockIdx.x * blockDim.x + threadIdx.x;
    size_t stride = (size_t)gridDim.x * blockDim.x;
    for (; i < n; i += stride) p[i] = v;
}

// ---------------------------------------------------------------------------
// GEMM1: h1[N,64] = x[N,128] @ W1[64,128]^T   (fp32 WMMA 16x16x4)
// block = 256 threads = 8 waves; wave (rg,cg): rows blockIdx*32+rg*16, cols cg*16
// ---------------------------------------------------------------------------
__global__ __launch_bounds__(256) void gemm1_wmma(const float* __restrict__ x,
                                                  const float* __restrict__ W,
                                                  float* __restrict__ h1, int N) {
    const int lane = threadIdx.x & 31;
    const int wave = threadIdx.x >> 5;
    const int row0 = blockIdx.x * 32 + (wave >> 2) * 16;
    const int col0 = (wave & 3) * 16;
    const int m    = lane & 15;
    const int kh   = (lane >> 4) * 2;   // 0 or 2 (K sub-pair per ISA A/B layout)

    int rA = row0 + m; if (rA >= N) rA = N - 1;   // clamp (exact for N=100000)
    const float* xrow = x + (size_t)rA * 128 + kh;
    const float* wrow = W + (size_t)(col0 + m) * 128 + kh;

    v8f acc = {};
    #pragma unroll 4
    for (int k = 0; k < 128; k += 4) {
        v2f a = *(const v2f*)(xrow + k);
        v2f b = *(const v2f*)(wrow + k);
        acc = __builtin_amdgcn_wmma_f32_16x16x4_f32(false, a, false, b,
                                                    (short)0, acc, false, false);
    }

    const int n16   = lane & 15;
    const int mBase = row0 + ((lane >> 4) << 3);
    float* out = h1 + (size_t)mBase * 64 + col0 + n16;
    if (row0 + 16 <= N) {
        // full tile: unguarded stores (common case; 16-lane-contiguous rows)
        #pragma unroll
        for (int i = 0; i < 8; ++i) out[(size_t)i * 64] = acc[i];
    } else {
        #pragma unroll
        for (int i = 0; i < 8; ++i)
            if (mBase + i < N) out[(size_t)i * 64] = acc[i];
    }
}

// ---------------------------------------------------------------------------
// GEMM2: h2[N,10] = x2[N,64] @ W2[10,64]^T  (N-dim padded 10 -> 16)
// block = 256 threads = 8 waves; each wave one 16-row tile -> 128 rows/block
// ---------------------------------------------------------------------------
__global__ __launch_bounds__(256) void gemm2_wmma(const float* __restrict__ x2,
                                                  const float* __restrict__ W2,
                                                  float* __restrict__ h2, int N) {
    const int lane = threadIdx.x & 31;
    const int wave = threadIdx.x >> 5;
    const int row0 = blockIdx.x * 128 + wave * 16;
    const int m    = lane & 15;
    const int kh   = (lane >> 4) * 2;

    int rA = row0 + m; if (rA >= N) rA = N - 1;
    const float* xrow = x2 + (size_t)rA * 64 + kh;
    const float* wrow = W2 + (size_t)m * 64 + kh;
    const bool bvalid = (m < 10);

    v8f acc = {};
    #pragma unroll 4
    for (int k = 0; k < 64; k += 4) {
        v2f a = *(const v2f*)(xrow + k);
        v2f b = {};
        if (bvalid) b = *(const v2f*)(wrow + k);
        acc = __builtin_amdgcn_wmma_f32_16x16x4_f32(false, a, false, b,
                                                    (short)0, acc, false, false);
    }

    const int n16   = lane & 15;
    const int mBase = row0 + ((lane >> 4) << 3);
    if (n16 < 10) {                       // uniform per lane: one divergence, not 8
        float* out = h2 + (size_t)mBase * 10 + n16;
        if (row0 + 16 <= N) {
            #pragma unroll
            for (int i = 0; i < 8; ++i) out[(size_t)i * 10] = acc[i];
        } else {
            #pragma unroll
            for (int i = 0; i < 8; ++i)
                if (mBase + i < N) out[(size_t)i * 10] = acc[i];
        }
    }
}

// ---------------------------------------------------------------------------
// layer-1 attention scores: s_src/s_dst [N,4]
// ---------------------------------------------------------------------------
__global__ void scores1_kernel(const float* __restrict__ h1,
                               const float* __restrict__ a_src,
                               const float* __restrict__ a_dst,
                               float* __restrict__ s_src, float* __restrict__ s_dst, int N) {
    int n = blockIdx.x * blockDim.x + threadIdx.x;
    if (n >= N) return;
    const float* hp = h1 + (size_t)n * 64;
    #pragma unroll
    for (int h = 0; h < 4; ++h) {
        float ss = 0.0f, sd = 0.0f;
        #pragma unroll
        for (int c = 0; c < 16; ++c) {
            float v = hp[h * 16 + c];
            ss += v * a_src[h * 16 + c];
            sd += v * a_dst[h * 16 + c];
        }
        s_src[n * 4 + h] = ss;
        s_dst[n * 4 + h] = sd;
    }
}

// layer-2 scores (heads=1, out=10)
__global__ void scores2_kernel(const float* __restrict__ h2,
                               const float* __restrict__ a_src,
                               const float* __restrict__ a_dst,
                               float* __restrict__ s_src, float* __restrict__ s_dst, int N) {
    int n = blockIdx.x * blockDim.x + threadIdx.x;
    if (n >= N) return;
    const float* hp = h2 + (size_t)n * 10;
    float ss = 0.0f, sd = 0.0f;
    #pragma unroll
    for (int c = 0; c < 10; ++c) {
        float v = hp[c];
        ss += v * a_src[c];
        sd += v * a_dst[c];
    }
    s_src[n] = ss;
    s_dst[n] = sd;
}

// ---------------------------------------------------------------------------
// edge passes, layer 1 (H=4)
// ---------------------------------------------------------------------------
__global__ void edge_max1(const long long* __restrict__ ei, int E0, int E,
                          const float* __restrict__ s_src, const float* __restrict__ s_dst,
                          float* __restrict__ m1) {
    int e = blockIdx.x * blockDim.x + threadIdx.x;
    if (e >= E) return;
    int s, d; edge_sd(ei, E0, e, &s, &d);
    #pragma unroll
    for (int h = 0; h < 4; ++h) {
        float v = lrelu(s_src[s * 4 + h] + s_dst[d * 4 + h]);
        atomicMaxF(&m1[d * 4 + h], v);
    }
}

__global__ void fixup_max(float* __restrict__ m, int n) {
    int i = blockIdx.x * blockDim.x + threadIdx.x;
    if (i < n) { float v = m[i]; if (!isfinite(v)) m[i] = 0.0f; }
}

__global__ void edge_sum1(const long long* __restrict__ ei, int E0, int E,
                          const float* __restrict__ s_src, const float* __restrict__ s_dst,
                          const float* __restrict__ m1, float* __restrict__ d1) {
    int e = blockIdx.x * blockDim.x + threadIdx.x;
    if (e >= E) return;
    int s, d; edge_sd(ei, E0, e, &s, &d);
    #pragma unroll
    for (int h = 0; h < 4; ++h) {
        float v = lrelu(s_src[s * 4 + h] + s_dst[d * 4 + h]);
        atomicAdd(&d1[d * 4 + h], expf(v - m1[d * 4 + h]));
    }
}

__global__ void edge_agg1(const long long* __restrict__ ei, int E0, int E,
                          const float* __restrict__ s_src, const float* __restrict__ s_dst,
                          const float* __restrict__ m1, const float* __restrict__ d1,
                          const float* __restrict__ h1, float* __restrict__ o1) {
    int e = blockIdx.x * blockDim.x + threadIdx.x;
    if (e >= E) return;
    int s, d; edge_sd(ei, E0, e, &s, &d);
    float alpha[4];
    #pragma unroll
    for (int h = 0; h < 4; ++h) {
        float v = lrelu(s_src[s * 4 + h] + s_dst[d * 4 + h]);
        alpha[h] = expf(v - m1[d * 4 + h]) / (d1[d * 4 + h] + 1e-16f);
    }
    const float4* hs = (const float4*)(h1 + (size_t)s * 64);
    float* op = o1 + (size_t)d * 64;
    #pragma unroll
    for (int h = 0; h < 4; ++h) {
        #pragma unroll
        for (int q = 0; q < 4; ++q) {
            float4 hv = hs[h * 4 + q];
            float* o = op + h * 16 + q * 4;
            atomicAdd(o + 0, alpha[h] * hv.x);
            atomicAdd(o + 1, alpha[h] * hv.y);
            atomicAdd(o + 2, alpha[h] * hv.z);
            atomicAdd(o + 3, alpha[h] * hv.w);
        }
    }
}

// bias + ELU in place on o1 -> becomes x2
__global__ void elu_bias(float* __restrict__ o1, const float* __restrict__ b1, size_t n64) {
    size_t i = (size_t)blockIdx.x * blockDim.x + threadIdx.x;
    if (i >= n64) return;
    float v = o1[i] + b1[i & 63];
    o1[i] = v > 0.0f ? v : expf(v) - 1.0f;
}

// ---------------------------------------------------------------------------
// edge passes, layer 2 (H=1, C=10)
// ---------------------------------------------------------------------------
__global__ void edge_max2(const long long* __restrict__ ei, int E0, int E,
                          const float* __restrict__ s_src, const float* __restrict__ s_dst,
                          float* __restrict__ m2) {
    int e = blockIdx.x * blockDim.x + threadIdx.x;
    if (e >= E) return;
    int s, d; edge_sd(ei, E0, e, &s, &d);
    atomicMaxF(&m2[d], lrelu(s_src[s] + s_dst[d]));
}

__global__ void edge_sum2(const long long* __restrict__ ei, int E0, int E,
                          const float* __restrict__ s_src, const float* __restrict__ s_dst,
                          const float* __restrict__ m2, float* __restrict__ d2) {
    int e = blockIdx.x * blockDim.x + threadIdx.x;
    if (e >= E) return;
    int s, d; edge_sd(ei, E0, e, &s, &d);
    float v = lrelu(s_src[s] + s_dst[d]);
    atomicAdd(&d2[d], expf(v - m2[d]));
}

__global__ void edge_agg2(const long long* __restrict__ ei, int E0, int E,
                          const float* __restrict__ s_src, const float* __restrict__ s_dst,
                          const float* __restrict__ m2, const float* __restrict__ d2,
                          const float* __restrict__ h2, float* __restrict__ o2) {
    int e = blockIdx.x * blockDim.x + threadIdx.x;
    if (e >= E) return;
    int s, d; edge_sd(ei, E0, e, &s, &d);
    float v = lrelu(s_src[s] + s_dst[d]);
    float alpha = expf(v - m2[d]) / (d2[d] + 1e-16f);
    const float* hp = h2 + (size_t)s * 10;
    float* op = o2 + (size_t)d * 10;
    #pragma unroll
    for (int c = 0; c < 10; ++c) atomicAdd(op + c, alpha * hp[c]);
}

// ---------------------------------------------------------------------------
// per-node bias + global mean pool accumulate
// ---------------------------------------------------------------------------
__global__ void pool_kernel(const float* __restrict__ o2, const float* __restrict__ b2,
                            const long long* __restrict__ batch,
                            float* __restrict__ pool, float* __restrict__ cnt, int N) {
    int n = blockIdx.x * blockDim.x + threadIdx.x;
    if (n >= N) return;
    int g = (int)batch[n];
    const float* op = o2 + (size_t)n * 10;
    #pragma unroll
    for (int c = 0; c < 10; ++c) atomicAdd(&pool[g * 10 + c], op[c] + b2[c]);
    atomicAdd(&cnt[g], 1.0f);
}

// mean + log_softmax over 10 classes, one thread per graph
__global__ void final_kernel(const float* __restrict__ pool, const float* __restrict__ cnt,
                             float* __restrict__ out, int G) {
    int g = blockIdx.x * blockDim.x + threadIdx.x;
    if (g >= G) return;
    float c = fmaxf(cnt[g], 1.0f);
    float v[10];
    float mx = -INFINITY;
    #pragma unroll
    for (int i = 0; i < 10; ++i) { v[i] = pool[g * 10 + i] / c; mx = fmaxf(mx, v[i]); }
    float s = 0.0f;
    #pragma unroll
    for (int i = 0; i < 10; ++i) s += expf(v[i] - mx);
    float lse = mx + logf(s);
    #pragma unroll
    for (int i = 0; i < 10; ++i) out[g * 10 + i] = v[i] - lse;
}

// ---------------------------------------------------------------------------
extern "C" void kernel_launch(void* const* d_in, const int* in_sizes, int n_in,
                              void* d_out, int out_size, void* d_ws, size_t ws_size,
                              hipStream_t stream) {
    (void)n_in; (void)ws_size;
    const float*     x      = (const float*)d_in[0];
    const long long* ei     = (const long long*)d_in[1];
    const long long* batch  = (const long long*)d_in[2];
    const float*     W1     = (const float*)d_in[3];
    const float*     a_src1 = (const float*)d_in[4];
    const float*     a_dst1 = (const float*)d_in[5];
    const float*     b1     = (const float*)d_in[6];
    const float*     W2     = (const float*)d_in[7];
    const float*     a_src2 = (const float*)d_in[8];
    const float*     a_dst2 = (const float*)d_in[9];
    const float*     b2     = (const float*)d_in[10];

    const int N  = in_sizes[0] / 128;
    const int E0 = in_sizes[1] / 2;
    const int E  = E0 + N;           // with self-loops
    const int G  = out_size / 10;

    // workspace carve (256B aligned slices)
    size_t off = 0;
    char* base = (char*)d_ws;
    auto alloc = [&](size_t nfloats) -> float* {
        float* p = (float*)(base + off);
        off += ((nfloats * sizeof(float)) + 255) & ~(size_t)255;
        return p;
    };
    float* h1     = alloc((size_t)N * 64);
    float* s_src1 = alloc((size_t)N * 4);
    float* s_dst1 = alloc((size_t)N * 4);
    float* m1     = alloc((size_t)N * 4);
    float* d1     = alloc((size_t)N * 4);
    float* o1     = alloc((size_t)N * 64);   // layer-1 aggregate, becomes x2
    float* h2     = alloc((size_t)N * 10);
    float* s_src2 = alloc((size_t)N);
    float* s_dst2 = alloc((size_t)N);
    float* m2     = alloc((size_t)N);
    float* d2     = alloc((size_t)N);
    float* o2     = alloc((size_t)N * 10);
    float* pool   = alloc((size_t)G * 10);
    float* cnt    = alloc((size_t)G);

    const int T = 256;
    auto blocks = [&](size_t n) { return (int)((n + T - 1) / T); };

    const float NEGINF = -INFINITY;
    // init (ws is poisoned; must re-init every call)
    fill_kernel<<<blocks((size_t)N * 64), T, 0, stream>>>(o1, 0.0f, (size_t)N * 64);
    fill_kernel<<<blocks((size_t)N * 4),  T, 0, stream>>>(m1, NEGINF, (size_t)N * 4);
    fill_kernel<<<blocks((size_t)N * 4),  T, 0, stream>>>(d1, 0.0f, (size_t)N * 4);
    fill_kernel<<<blocks((size_t)N),      T, 0, stream>>>(m2, NEGINF, (size_t)N);
    fill_kernel<<<blocks((size_t)N),      T, 0, stream>>>(d2, 0.0f, (size_t)N);
    fill_kernel<<<blocks((size_t)N * 10), T, 0, stream>>>(o2, 0.0f, (size_t)N * 10);
    fill_kernel<<<blocks((size_t)G * 10), T, 0, stream>>>(pool, 0.0f, (size_t)G * 10);
    fill_kernel<<<blocks((size_t)G),      T, 0, stream>>>(cnt, 0.0f, (size_t)G);

    // layer 1
    gemm1_wmma<<<(N + 31) / 32, 256, 0, stream>>>(x, W1, h1, N);
    scores1_kernel<<<blocks(N), T, 0, stream>>>(h1, a_src1, a_dst1, s_src1, s_dst1, N);
    edge_max1<<<blocks(E), T, 0, stream>>>(ei, E0, E, s_src1, s_dst1, m1);
    fixup_max<<<blocks((size_t)N * 4), T, 0, stream>>>(m1, N * 4);
    edge_sum1<<<blocks(E), T, 0, stream>>>(ei, E0, E, s_src1, s_dst1, m1, d1);
    edge_agg1<<<blocks(E), T, 0, stream>>>(ei, E0, E, s_src1, s_dst1, m1, d1, h1, o1);
    elu_bias<<<blocks((size_t)N * 64), T, 0, stream>>>(o1, b1, (size_t)N * 64);

    // layer 2
    gemm2_wmma<<<(N + 127) / 128, 256, 0, stream>>>(o1, W2, h2, N);
    scores2_kernel<<<blocks(N), T, 0, stream>>>(h2, a_src2, a_dst2, s_src2, s_dst2, N);
    edge_max2<<<blocks(E), T, 0, stream>>>(ei, E0, E, s_src2, s_dst2, m2);
    fixup_max<<<blocks(N), T, 0, stream>>>(m2, N);
    edge_sum2<<<blocks(E), T, 0, stream>>>(ei, E0, E, s_src2, s_dst2, m2, d2);
    edge_agg2<<<blocks(E), T, 0, stream>>>(ei, E0, E, s_src2, s_dst2, m2, d2, h2, o2);

    // pool + log_softmax
    pool_kernel<<<blocks(N), T, 0, stream>>>(o2, b2, batch, pool, cnt, N);
    final_kernel<<<1, ((G + 63) / 64) * 64, 0, stream>>>(pool, cnt, (float*)d_out, G);
}